// RoGPELinearNodeEncoder_45045617000786
// MI455X (gfx1250) — compile-verified
//
#include <hip/hip_runtime.h>

typedef __attribute__((ext_vector_type(2))) float v2f;
typedef __attribute__((ext_vector_type(8))) float v8f;

// ---------------------------------------------------------------- utilities
__global__ __launch_bounds__(256) void zero_u32_kernel(unsigned* __restrict__ p, int n) {
    int i = blockIdx.x * blockDim.x + threadIdx.x;
    if (i < n) p[i] = 0u;
}

// ------------------------------------------------- fused GEMM + bias + PReLU
// Y[N,O] = prelu(X[N,D] @ W[O,D]^T (+ bias), slope)
// Register-blocked: one wave -> 2x2 grid of 16x16 tiles (32x32 output) via
// V_WMMA_F32_16X16X4_F32. Per K-step: 2 A-frags + 2 B-frags -> 4 WMMAs
// (1 load : 1 WMMA instead of 2:1), 4 independent accumulator chains.
// Block = 256 threads = 8 waves as 4(M) x 2(N) -> 128x64 macro-tile.
__global__ __launch_bounds__(256) void gemm_prelu_wmma(
    const float* __restrict__ X, const float* __restrict__ W,
    const float* __restrict__ bias, const float* __restrict__ slope_p,
    float* __restrict__ Y, int D, int O)
{
    const int lane = threadIdx.x & 31;
    const int wave = threadIdx.x >> 5;
    const int wm = wave & 3;                 // wave M position (0..3)
    const int wn = wave >> 2;                // wave N position (0..1)
    const int m0 = blockIdx.y * 128 + wm * 32;
    const int n0 = blockIdx.x * 64 + wn * 32;
    const int row = lane & 15;               // M idx (A) / N idx (B,C)
    const int kh  = (lane >> 4) << 1;        // K sub-offset: 0 or 2

    const float* Ap0 = X + (size_t)(m0 + row) * D + kh;
    const float* Ap1 = X + (size_t)(m0 + 16 + row) * D + kh;
    const float* Bp0 = W + (size_t)(n0 + row) * D + kh;        // B = W^T frag
    const float* Bp1 = W + (size_t)(n0 + 16 + row) * D + kh;

    v8f acc00 = {}, acc01 = {}, acc10 = {}, acc11 = {};
#pragma unroll 4
    for (int k0 = 0; k0 < D; k0 += 4) {
        v2f a0 = *(const v2f*)(Ap0 + k0);
        v2f a1 = *(const v2f*)(Ap1 + k0);
        v2f b0 = *(const v2f*)(Bp0 + k0);
        v2f b1 = *(const v2f*)(Bp1 + k0);
        acc00 = __builtin_amdgcn_wmma_f32_16x16x4_f32(
            false, a0, false, b0, (short)0, acc00, false, false);
        acc01 = __builtin_amdgcn_wmma_f32_16x16x4_f32(
            false, a0, false, b1, (short)0, acc01, false, false);
        acc10 = __builtin_amdgcn_wmma_f32_16x16x4_f32(
            false, a1, false, b0, (short)0, acc10, false, false);
        acc11 = __builtin_amdgcn_wmma_f32_16x16x4_f32(
            false, a1, false, b1, (short)0, acc11, false, false);
    }

    const float slope = slope_p[0];
    const float bv0 = bias ? bias[n0 + row] : 0.0f;
    const float bv1 = bias ? bias[n0 + 16 + row] : 0.0f;
    const int mb = m0 + ((lane >> 4) << 3);  // M = r + 8*(lane>=16) within tile
    const int c0 = n0 + row;
    const int c1 = n0 + 16 + row;
#pragma unroll
    for (int r = 0; r < 8; ++r) {
        float v00 = acc00[r] + bv0;
        float v01 = acc01[r] + bv1;
        float v10 = acc10[r] + bv0;
        float v11 = acc11[r] + bv1;
        v00 = (v00 >= 0.0f) ? v00 : slope * v00;
        v01 = (v01 >= 0.0f) ? v01 : slope * v01;
        v10 = (v10 >= 0.0f) ? v10 : slope * v10;
        v11 = (v11 >= 0.0f) ? v11 : slope * v11;
        Y[(size_t)(mb + r) * O + c0] = v00;
        Y[(size_t)(mb + r) * O + c1] = v01;
        Y[(size_t)(mb + 16 + r) * O + c0] = v10;
        Y[(size_t)(mb + 16 + r) * O + c1] = v11;
    }
}

// ------------------------------------------------- output layer: angles = X @ Wo^T + bo
__global__ __launch_bounds__(256) void out_matvec(
    const float* __restrict__ X, const float* __restrict__ Wo,
    const float* __restrict__ bo, float* __restrict__ ang, int N, int D)
{
    int n = blockIdx.x * 8 + (threadIdx.x >> 5);   // one node per wave
    int lane = threadIdx.x & 31;
    if (n >= N) return;
    const float* xp = X + (size_t)n * D;
    float s = 0.0f;
    for (int d = lane; d < D; d += 32) s += xp[d] * Wo[d];
#pragma unroll
    for (int off = 16; off > 0; off >>= 1) s += __shfl_down(s, off, 32);
    if (lane == 0) ang[n] = s + bo[0];
}

// ------------------------------------------------- adjacency bitset: A[src][dst] = 1
__global__ __launch_bounds__(256) void build_adj(
    const int* __restrict__ ei, int E, unsigned* __restrict__ A, int words)
{
    int e = blockIdx.x * blockDim.x + threadIdx.x;
    if (e >= E) return;
    int s = ei[e];
    int d = ei[E + e];
    atomicOr(&A[(size_t)s * words + (d >> 5)], 1u << (d & 31));
}

// ------------------------------------------------- hop 1: tmp[src] += ang[dst] (raw dup edges)
__global__ __launch_bounds__(256) void hop1_scatter(
    const int* __restrict__ ei, int E, const float* __restrict__ ang,
    float* __restrict__ tmp)
{
    int e = blockIdx.x * blockDim.x + threadIdx.x;
    if (e >= E) return;
    int s = ei[e];
    int d = ei[E + e];
    atomicAdd(&tmp[s], ang[d]);
}

__global__ __launch_bounds__(256) void axpy_combine(
    const float* __restrict__ a, const float* __restrict__ t,
    float* __restrict__ out, float scale, int N)
{
    int n = blockIdx.x * blockDim.x + threadIdx.x;
    if (n < N) out[n] = a[n] + scale * t[n];
}

// ------------------------------------------------- boolean matrix power + SpMV
// RowOut(i) = OR_{k in Ain_row(i)} Abase_row(k);  vout[i] = vin[i] + scale * sum_{j in RowOut} vin[j]
// Block = `words` threads (one u32 word-column per thread); grid = N rows.
__global__ __launch_bounds__(128) void hop_kernel(
    const unsigned* __restrict__ Ain, const unsigned* __restrict__ Abase,
    const float* __restrict__ vin, float* __restrict__ vout,
    unsigned* __restrict__ Aout, float scale, int words)
{
    const int i = blockIdx.x;
    const int t = threadIdx.x;
    const unsigned* rin = Ain + (size_t)i * words;

    unsigned accw = 0u;
    for (int j = 0; j < words; ++j) {
        unsigned m = rin[j];                 // same word across lanes: broadcast
        while (m) {
            int b = __ffs(m) - 1;
            m &= m - 1u;
            int k = (j << 5) + b;
            accw |= Abase[(size_t)k * words + t];   // coalesced 512B row read
        }
    }
    if (Aout) Aout[(size_t)i * words + t] = accw;

    // sparse dot: sum vin over set bits of this word
    float s = 0.0f;
    unsigned m = accw;
    const int base = t << 5;
    while (m) {
        int b = __ffs(m) - 1;
        m &= m - 1u;
        s += vin[base + b];
    }

    __shared__ float red[256];
    red[t] = s;
    __syncthreads();
    for (int off = blockDim.x >> 1; off > 0; off >>= 1) {
        if (t < off) red[t] += red[t + off];
        __syncthreads();
    }
    if (t == 0) vout[i] = vin[i] + scale * red[0];
}

// ---------------------------------------------------------------- launcher
extern "C" void kernel_launch(void* const* d_in, const int* in_sizes, int n_in,
                              void* d_out, int out_size, void* d_ws, size_t ws_size,
                              hipStream_t stream)
{
    const float* coeffs = (const float*)d_in[0];
    const int*   eidx   = (const int*)d_in[1];
    const float* W0 = (const float*)d_in[2];
    const float* b0 = (const float*)d_in[3];
    const float* a0 = (const float*)d_in[4];
    const float* W1 = (const float*)d_in[5];
    const float* a1 = (const float*)d_in[6];
    const float* W2 = (const float*)d_in[7];
    const float* a2 = (const float*)d_in[8];
    const float* Wo = (const float*)d_in[9];
    const float* bo = (const float*)d_in[10];
    float* out = (float*)d_out;

    const int D = in_sizes[3];          // 512 (b0 length)
    const int N = in_sizes[0] / D;      // 4096
    const int E = in_sizes[1] / 2;      // 131072
    const int words = (N + 31) / 32;    // 128

    char* ws = (char*)d_ws;
    float* X1 = (float*)ws;                 ws += (size_t)N * D * sizeof(float);
    float* X2 = (float*)ws;                 ws += (size_t)N * D * sizeof(float);
    float* ang = (float*)ws;                ws += (size_t)N * sizeof(float);
    float* tmp = (float*)ws;                ws += (size_t)N * sizeof(float);
    float* v1  = (float*)ws;                ws += (size_t)N * sizeof(float);
    float* v2  = (float*)ws;                ws += (size_t)N * sizeof(float);
    unsigned* A   = (unsigned*)ws;          ws += (size_t)N * words * sizeof(unsigned);
    unsigned* A2m = (unsigned*)ws;          ws += (size_t)N * words * sizeof(unsigned);

    // zero adjacency bitset + hop-1 accumulator
    {
        int nw = N * words;
        zero_u32_kernel<<<(nw + 255) / 256, 256, 0, stream>>>(A, nw);
        zero_u32_kernel<<<(N + 255) / 256, 256, 0, stream>>>((unsigned*)tmp, N);
    }

    // MLP: three WMMA GEMMs (fused bias + PReLU), ping-pong X1/X2
    dim3 ggrid(D / 64, N / 128);
    gemm_prelu_wmma<<<ggrid, 256, 0, stream>>>(coeffs, W0, b0,      a0, X1, D, D);
    gemm_prelu_wmma<<<ggrid, 256, 0, stream>>>(X1,     W1, nullptr, a1, X2, D, D);
    gemm_prelu_wmma<<<ggrid, 256, 0, stream>>>(X2,     W2, nullptr, a2, X1, D, D);

    // angles = X1 @ Wo^T + bo
    out_matvec<<<(N + 7) / 8, 256, 0, stream>>>(X1, Wo, bo, ang, N, D);

    // adjacency bitset (dedup by construction)
    build_adj<<<(E + 255) / 256, 256, 0, stream>>>(eidx, E, A, words);

    // hop 1: raw (duplicated) edge list, a1 = ang + 0.5^2 * segsum
    hop1_scatter<<<(E + 255) / 256, 256, 0, stream>>>(eidx, E, ang, tmp);
    axpy_combine<<<(N + 255) / 256, 256, 0, stream>>>(ang, tmp, v1, 0.25f, N);

    // hop 2: A2 = bool(A@A); a2 = a1 + (1/3)^2 * A2@a1
    hop_kernel<<<N, words, 0, stream>>>(A, A, v1, v2, A2m, 1.0f / 9.0f, words);
    // hop 3: A3 = bool(A2@A); out = a2 + (1/4)^2 * A3@a2 (A3 not materialized)
    hop_kernel<<<N, words, 0, stream>>>(A2m, A, v2, out, nullptr, 1.0f / 16.0f, words);
}